// RELGLayer_29712583754016
// MI455X (gfx1250) — compile-verified
//
#include <hip/hip_runtime.h>
#include <hip/hip_bf16.h>
#include <math.h>

// ---------------------------------------------------------------------------
// RELG layer for MI455X (gfx1250, wave32).
// GEMMs run on the matrix pipe via V_WMMA_F32_16X16X4_F32 (fp32-exact).
// Fragment layouts per CDNA5 ISA 7.12.2:
//   A (16x4 f32):  lanes 0-15: {A[m][k],A[m][k+1]}, lanes16-31: {A[m][k+2],A[m][k+3]}
//   B (4x16 f32):  lanes 0-15: {B[k][n],B[k+1][n]}, lanes16-31: {B[k+2][n],B[k+3][n]}
//   C/D (16x16):   vgpr j, lane L: M = j + 8*(L/16), N = L%16
// A and B staged in LDS (rows padded to 132 floats = 528B, 16B-aligned, so the
// K-strided A reads spread across banks). Each wave owns a 32x32 register tile
// (4 accumulators; 2 A x 2 B fragments per k-step -> 4 WMMA, halving DS traffic
// per WMMA). Each 256-thread block covers a 64-row x 128-col tile; total LDS
// ~101KB of the 320KB/WGP.
// ---------------------------------------------------------------------------

typedef __attribute__((ext_vector_type(2))) float v2f;
typedef __attribute__((ext_vector_type(8))) float v8f;

#define D128 128
#define LDSP 132        // padded LDS row length (floats)
#define NCOLS 1408      // 11 * 128 : [Ah | Bh | Eh | HW_0..HW_7]
#define OFF_A 0
#define OFF_B 128
#define OFF_E 256
#define OFF_MSG 384

__device__ __forceinline__ v8f wmma4(v2f a, v2f b, v8f c) {
  return __builtin_amdgcn_wmma_f32_16x16x4_f32(false, a, false, b, (short)0, c,
                                               false, false);
}

// Wave tile: rows [arow0, arow0+32) x cols [c0, c0+32) of the block tile.
__device__ __forceinline__ void wmma_quad_k128(const float (*As)[LDSP], int arow0,
                                               const float (*Bs)[LDSP], int c0,
                                               int half, int l16,
                                               v8f& q00, v8f& q01,
                                               v8f& q10, v8f& q11) {
#pragma unroll
  for (int k = 0; k < 128; k += 4) {
    const int kk = k + 2 * half;
    v2f a0, a1, b0, b1;
    a0.x = As[arow0 + l16][kk];
    a0.y = As[arow0 + l16][kk + 1];
    a1.x = As[arow0 + 16 + l16][kk];
    a1.y = As[arow0 + 16 + l16][kk + 1];
    b0.x = Bs[kk][c0 + l16];
    b0.y = Bs[kk + 1][c0 + l16];
    b1.x = Bs[kk][c0 + 16 + l16];
    b1.y = Bs[kk + 1][c0 + 16 + l16];
    q00 = wmma4(a0, b0, q00);
    q01 = wmma4(a0, b1, q01);
    q10 = wmma4(a1, b0, q10);
    q11 = wmma4(a1, b1, q11);
  }
}

// ---------------------------------------------------------------------------
// 0) zero scratch accumulators
__global__ void zero_ws(float* __restrict__ p, long n) {
  long i = (long)blockIdx.x * blockDim.x + threadIdx.x;
  if (i < n) p[i] = 0.f;
}

// ---------------------------------------------------------------------------
// 1) Wcat[k][o] (128 x 1408) and bcat[o]; folds basis decomposition.
__global__ void build_weights(const float* __restrict__ Aw, const float* __restrict__ Bw,
                              const float* __restrict__ Ew, const float* __restrict__ weight,
                              const float* __restrict__ w_comp,
                              const float* __restrict__ Ab, const float* __restrict__ Bb,
                              const float* __restrict__ Eb,
                              float* __restrict__ Wcat, float* __restrict__ bcat) {
  int idx = blockIdx.x * blockDim.x + threadIdx.x;
  if (idx < D128 * NCOLS) {
    int k = idx / NCOLS, o = idx % NCOLS;
    float v;
    if (o < 128)       v = Aw[k * 128 + o];
    else if (o < 256)  v = Bw[k * 128 + (o - 128)];
    else if (o < 384)  v = Ew[k * 128 + (o - 256)];
    else {
      int r = (o - 384) >> 7, oo = (o - 384) & 127;
      float s = 0.f;
#pragma unroll
      for (int b = 0; b < 4; b++)
        s += w_comp[r * 4 + b] * weight[((size_t)b * 128 + k) * 128 + oo];
      v = s;
    }
    Wcat[idx] = v;
  }
  if (idx < NCOLS) {
    bcat[idx] = (idx < 128)   ? Ab[idx]
              : (idx < 256)   ? Bb[idx - 128]
              : (idx < 384)   ? Eb[idx - 256]
                              : 0.f;
  }
}

// ---------------------------------------------------------------------------
// shared staging helpers (256 threads)
__device__ __forceinline__ void stage_B(float (*Bs)[LDSP], const float* __restrict__ B,
                                        int ldb, int tid) {
  // 128 x 128 floats = 4096 float4, 16 per thread, coalesced b128 loads
  for (int i4 = tid; i4 < 128 * 32; i4 += 256) {
    int row = i4 >> 5, c4 = (i4 & 31) << 2;
    const float4 v = *reinterpret_cast<const float4*>(B + (size_t)row * ldb + c4);
    *reinterpret_cast<float4*>(&Bs[row][c4]) = v;
  }
}

__device__ __forceinline__ void stage_A64(float (*As)[LDSP], const float* __restrict__ A,
                                          int m0, int rows, int tid) {
  // 64 x 128 floats = 2048 float4, 8 per thread
  for (int i4 = tid; i4 < 64 * 32; i4 += 256) {
    int r = i4 >> 5, c4 = (i4 & 31) << 2;
    int gm = m0 + r;
    float4 v = make_float4(0.f, 0.f, 0.f, 0.f);
    if (gm < rows) {
      v = *reinterpret_cast<const float4*>(A + (size_t)gm * D128 + c4);
      // pull the next block's rows toward GL2 (global_prefetch_b8)
      if (gm + 64 < rows)
        __builtin_prefetch(A + (size_t)(gm + 64) * D128 + c4, 0, 1);
    }
    *reinterpret_cast<float4*>(&As[r][c4]) = v;
  }
}

// ---------------------------------------------------------------------------
// 2) node GEMM: out_node[N x 1408] = h @ Wcat + bcat
//    block = 64 rows x 128 cols, 8 waves, wave w -> 32x32 register tile
__global__ void __launch_bounds__(256)
node_gemm(const float* __restrict__ h, const float* __restrict__ Wcat,
          const float* __restrict__ bcat, float* __restrict__ out, int N) {
  __shared__ float As[64][LDSP];
  __shared__ float Bs[128][LDSP];
  const int m0 = blockIdx.x * 64;
  const int n0 = blockIdx.y * 128;
  const int tid = threadIdx.x;
  stage_A64(As, h, m0, N, tid);
  stage_B(Bs, Wcat + n0, NCOLS, tid);
  __syncthreads();
  const int wave = tid >> 5, lane = tid & 31;
  const int half = lane >> 4, l16 = lane & 15;
  const int arow0 = (wave >> 2) * 32;      // 0 or 32
  const int c0 = (wave & 3) * 32;          // 0,32,64,96
  v8f q00 = {}, q01 = {}, q10 = {}, q11 = {};
  wmma_quad_k128(As, arow0, Bs, c0, half, l16, q00, q01, q10, q11);
  const float bias0 = bcat[n0 + c0 + l16];
  const float bias1 = bcat[n0 + c0 + 16 + l16];
#pragma unroll
  for (int mi = 0; mi < 2; mi++) {
    const v8f& qa = mi ? q10 : q00;
    const v8f& qb = mi ? q11 : q01;
#pragma unroll
    for (int j = 0; j < 8; j++) {
      int gm = m0 + arow0 + mi * 16 + j + 8 * half;
      if (gm < N) {
        float* orow = out + (size_t)gm * NCOLS + n0 + c0 + l16;
        orow[0]  = qa[j] + bias0;
        orow[16] = qb[j] + bias1;
      }
    }
  }
}

// ---------------------------------------------------------------------------
// 3) fused edge kernel: e_ij = e@Cw + Cb + msg[src,etype] + Eh[dst]
//    writes graph-normed e_ij to d_out, scatters sigma*Bh[src], sigma via atomics
__global__ void __launch_bounds__(256)
edge_gemm_fused(const float* __restrict__ e, const float* __restrict__ Cw,
                const float* __restrict__ Cb,
                const int* __restrict__ src, const int* __restrict__ dst,
                const int* __restrict__ etype,
                const float* __restrict__ out_node,
                float* __restrict__ num, float* __restrict__ den,
                float* __restrict__ e_out, int E, float invSqrtE) {
  __shared__ float As[64][LDSP];
  __shared__ float Bs[128][LDSP];
  __shared__ int sSrc[64], sDst[64], sEt[64];
  const int m0 = blockIdx.x * 64;
  const int tid = threadIdx.x;
  stage_A64(As, e, m0, E, tid);
  stage_B(Bs, Cw, D128, tid);
  if (tid < 64) {
    int gm = m0 + tid;
    int cl = (gm < E) ? gm : (E - 1);
    sSrc[tid] = src[cl];
    sDst[tid] = dst[cl];
    sEt[tid]  = etype[cl];
  }
  __syncthreads();
  const int wave = tid >> 5, lane = tid & 31;
  const int half = lane >> 4, l16 = lane & 15;
  const int arow0 = (wave >> 2) * 32;
  const int c0 = (wave & 3) * 32;
  v8f q00 = {}, q01 = {}, q10 = {}, q11 = {};
  wmma_quad_k128(As, arow0, Bs, c0, half, l16, q00, q01, q10, q11);
#pragma unroll
  for (int mi = 0; mi < 2; mi++) {
#pragma unroll
    for (int sub = 0; sub < 2; sub++) {
      const v8f& q = mi ? (sub ? q11 : q10) : (sub ? q01 : q00);
      const int n = c0 + sub * 16 + l16;   // output feature 0..127
      const float bias = Cb[n];
#pragma unroll
      for (int j = 0; j < 8; j++) {
        const int M = arow0 + mi * 16 + j + 8 * half;
        const int gm = m0 + M;
        if (gm < E) {
          const int sI = sSrc[M], dI = sDst[M], et = sEt[M];
          const float* srow = out_node + (size_t)sI * NCOLS;
          const float* drow = out_node + (size_t)dI * NCOLS;
          float v = q[j] + bias + srow[OFF_MSG + et * 128 + n] + drow[OFF_E + n];
          float sg = 1.f / (1.f + __expf(-v));
          e_out[(size_t)gm * D128 + n] = v * invSqrtE;   // graph-normed e_ij
          atomicAdd(&num[(size_t)dI * D128 + n], sg * srow[OFF_B + n]);
          atomicAdd(&den[(size_t)dI * D128 + n], sg);
        }
      }
    }
  }
}

// ---------------------------------------------------------------------------
// 4) node finalize + BN partial stats. den[n*128] > 0  <=>  deg > 0 (sigmoid>0).
__global__ void __launch_bounds__(128)
node_finalize(const float* __restrict__ out_node, const float* __restrict__ num,
              const float* __restrict__ den, const float* __restrict__ h_in,
              float* __restrict__ h_out, float* __restrict__ stats,
              int N, float invSqrtN) {
  const int col = threadIdx.x;
  const int r0 = blockIdx.x * 64;
  float s = 0.f, s2 = 0.f;
  for (int i = 0; i < 64; i++) {
    int nrow = r0 + i;
    if (nrow >= N) break;
    float d0 = den[(size_t)nrow * D128];
    float v;
    if (d0 > 0.f) {
      float dn = den[(size_t)nrow * D128 + col];
      v = out_node[(size_t)nrow * NCOLS + OFF_A + col] +
          num[(size_t)nrow * D128 + col] / (dn + 1e-6f);
    } else {
      v = h_in[(size_t)nrow * D128 + col];
    }
    v *= invSqrtN;
    h_out[(size_t)nrow * D128 + col] = v;
    s += v;
    s2 += v * v;
  }
  atomicAdd(&stats[col], s);
  atomicAdd(&stats[128 + col], s2);
}

// ---------------------------------------------------------------------------
// 5) column stats over already-scaled edge output
__global__ void __launch_bounds__(128)
col_stats(const float* __restrict__ x, float* __restrict__ sum,
          float* __restrict__ sumsq, int rows) {
  const int col = threadIdx.x;
  const int r0 = blockIdx.x * 256;
  float s = 0.f, s2 = 0.f;
  for (int i = 0; i < 256; i++) {
    int r = r0 + i;
    if (r >= rows) break;
    float v = x[(size_t)r * D128 + col];
    s += v;
    s2 += v * v;
  }
  atomicAdd(&sum[col], s);
  atomicAdd(&sumsq[col], s2);
}

// ---------------------------------------------------------------------------
// 6) BN (batch stats) + ReLU + residual, in place on d_out region
__global__ void bn_relu_res(const float* __restrict__ x_res, float* __restrict__ xio,
                            const float* __restrict__ sum, const float* __restrict__ sumsq,
                            const float* __restrict__ gamma, const float* __restrict__ beta,
                            long total, float inv_cnt) {
  long idx = (long)blockIdx.x * blockDim.x + threadIdx.x;
  if (idx >= total) return;
  int col = (int)(idx & 127);
  float mu = sum[col] * inv_cnt;
  float var = sumsq[col] * inv_cnt - mu * mu;
  float v = (xio[idx] - mu) * rsqrtf(var + 1e-5f) * gamma[col] + beta[col];
  v = fmaxf(v, 0.f);
  xio[idx] = x_res[idx] + v;
}

// ---------------------------------------------------------------------------
extern "C" void kernel_launch(void* const* d_in, const int* in_sizes, int n_in,
                              void* d_out, int out_size, void* d_ws, size_t ws_size,
                              hipStream_t stream) {
  const float* h      = (const float*)d_in[0];
  const float* e      = (const float*)d_in[1];
  const int*   src    = (const int*)d_in[2];
  const int*   dst    = (const int*)d_in[3];
  const int*   etype  = (const int*)d_in[4];
  const float* Aw     = (const float*)d_in[5];
  const float* Ab     = (const float*)d_in[6];
  const float* Bw     = (const float*)d_in[7];
  const float* Bb     = (const float*)d_in[8];
  const float* Cw     = (const float*)d_in[9];
  const float* Cb     = (const float*)d_in[10];
  const float* Ew     = (const float*)d_in[11];
  const float* Eb     = (const float*)d_in[12];
  const float* weight = (const float*)d_in[13];
  const float* w_comp = (const float*)d_in[14];
  const float* gam_h  = (const float*)d_in[15];
  const float* bet_h  = (const float*)d_in[16];
  const float* gam_e  = (const float*)d_in[17];
  const float* bet_e  = (const float*)d_in[18];

  const int N = in_sizes[0] / D128;
  const int E = in_sizes[1] / D128;

  // workspace layout (floats)
  float* ws      = (float*)d_ws;
  float* num     = ws;                               // N*128
  float* den     = num + (size_t)N * D128;           // N*128
  float* stats   = den + (size_t)N * D128;           // 512: hSum,hSq,eSum,eSq
  float* bcat    = stats + 512;                      // 1408
  float* Wcat    = bcat + NCOLS;                     // 128*1408
  float* outnode = Wcat + (size_t)D128 * NCOLS;      // N*1408

  float* h_out = (float*)d_out;
  float* e_out = h_out + (size_t)N * D128;

  const float invSqrtN = 1.f / sqrtf((float)N);
  const float invSqrtE = 1.f / sqrtf((float)E);

  // 0) zero accumulators + stats
  {
    long cnt = 2L * N * D128 + 512;
    zero_ws<<<dim3((unsigned)((cnt + 255) / 256)), 256, 0, stream>>>(num, cnt);
  }
  // 1) fold weights
  {
    int tot = D128 * NCOLS;
    build_weights<<<dim3((tot + 255) / 256), 256, 0, stream>>>(
        Aw, Bw, Ew, weight, w_comp, Ab, Bb, Eb, Wcat, bcat);
  }
  // 2) node GEMM -> [Ah|Bh|Eh|HW_r]
  {
    dim3 grid((N + 63) / 64, NCOLS / 128);
    node_gemm<<<grid, 256, 0, stream>>>(h, Wcat, bcat, outnode, N);
  }
  // 3) fused edge GEMM + gathers + sigmoid + scatter-atomics
  {
    dim3 grid((E + 63) / 64);
    edge_gemm_fused<<<grid, 256, 0, stream>>>(e, Cw, Cb, src, dst, etype, outnode,
                                              num, den, e_out, E, invSqrtE);
  }
  // 4) node aggregate + graph norm + h BN stats
  node_finalize<<<dim3((N + 63) / 64), 128, 0, stream>>>(
      outnode, num, den, h, h_out, stats, N, invSqrtN);
  // 5) e BN stats
  col_stats<<<dim3((E + 255) / 256), 128, 0, stream>>>(e_out, stats + 256, stats + 384, E);
  // 6) BN + ReLU + residual
  {
    long totH = (long)N * D128;
    bn_relu_res<<<dim3((unsigned)((totH + 255) / 256)), 256, 0, stream>>>(
        h, h_out, stats, stats + 128, gam_h, bet_h, totH, 1.f / (float)N);
    long totE = (long)E * D128;
    bn_relu_res<<<dim3((unsigned)((totE + 255) / 256)), 256, 0, stream>>>(
        e, e_out, stats + 256, stats + 384, gam_e, bet_e, totE, 1.f / (float)E);
  }
}